// DynaResidualBlock_48447231099375
// MI455X (gfx1250) — compile-verified
//
#include <hip/hip_runtime.h>
#include <hip/hip_bf16.h>

typedef float v2f __attribute__((ext_vector_type(2)));
typedef float v8f __attribute__((ext_vector_type(8)));

__device__ __forceinline__ v8f wmma4(v2f a, v2f b, v8f c) {
    // D = A(16x4 f32) * B(4x16 f32) + C(16x16 f32)
    return __builtin_amdgcn_wmma_f32_16x16x4_f32(
        /*neg_a=*/false, a, /*neg_b=*/false, b,
        /*c_mod=*/(short)0, c, /*reuse_a=*/false, /*reuse_b=*/false);
}

// ---------------------------------------------------------------------------
// Problem constants
// ---------------------------------------------------------------------------
#define NB    32
#define LATN  256
#define FIN   32
#define FOUT  32
#define FH    64
#define HWPX  16384          // 128*128
#define KTOT  9408
// slices inside the generated parameter vector (per sample)
#define OFF_KIN   0          // 64*32  = 2048
#define OFF_KMID  2048       // 64*64  = 4096
#define OFF_KOUT  6144       // 32*64  = 2048
#define OFF_KSH   8192       // 32*32  = 1024
#define OFF_BIN   9216       // 64
#define OFF_BMID  9280       // 64
#define OFF_BOUT  9344       // 32
#define OFF_BSH   9376       // 32

#define STG_STRIDE 65        // 64 + 1 pad -> conflict-free LDS stage
#define STG_PER_WAVE (16 * STG_STRIDE)   // 1040 floats

// ---------------------------------------------------------------------------
// Hypernetwork GEMM:  C[32,N] = A[32,K] @ W[K,N] + bias[N]
// One wave -> two 16-row tiles x one 16-col tile. 128 threads = 4 waves.
// All strided accesses use 32-bit element offsets from a uniform base so the
// backend can use GVS addressing (saddr + 32-bit voffset, scale_offset).
// ---------------------------------------------------------------------------
__global__ void __launch_bounds__(128)
hyper_gemm(const float* __restrict__ A, const float* __restrict__ W,
           const float* __restrict__ bias, float* __restrict__ C,
           int K, int N) {
    const int wave = threadIdx.x >> 5;
    const int lane = threadIdx.x & 31;
    const int n    = lane & 15;
    const int half = lane >> 4;
    const int kk   = half * 2;
    const int col0 = (blockIdx.x * 4 + wave) * 16;

    const float bv = bias[col0 + n];
    v8f acc0, acc1;
#pragma unroll
    for (int r = 0; r < 8; ++r) { acc0[r] = bv; acc1[r] = bv; }

    // per-lane constant parts
    const unsigned uN   = (unsigned)N;
    unsigned       wOff = (unsigned)kk * uN + (unsigned)(col0 + n); // W[k0][col]
    const float* __restrict__ A0 = A + (size_t)n * K + kk;          // row n
    const float* __restrict__ A1 = A + (size_t)(n + 16) * K + kk;   // row n+16

#pragma unroll 8
    for (int kb = 0; kb < K; kb += 4) {
        v2f bf; bf.x = W[wOff];
                bf.y = W[wOff + uN];
        v2f a0; a0.x = A0[kb]; a0.y = A0[kb + 1];
        v2f a1; a1.x = A1[kb]; a1.y = A1[kb + 1];
        acc0 = wmma4(a0, bf, acc0);
        acc1 = wmma4(a1, bf, acc1);
        wOff += 4u * uN;
    }

    unsigned cOff = (unsigned)(half * 8) * uN + (unsigned)(col0 + n);
#pragma unroll
    for (int r = 0; r < 8; ++r) {
        C[cOff + (unsigned)r * uN]            = acc0[r];
        C[cOff + (unsigned)(r + 16) * uN]     = acc1[r];
    }
}

// ---------------------------------------------------------------------------
// Fused dynamic residual block (per-sample 1x1 conv chain + shortcut).
// grid = (HW / 64, B); block = 128 (4 waves, 16 cols each).
// LDS: 9408 generated params + 4 per-wave transpose stages.
// ---------------------------------------------------------------------------
__global__ void __launch_bounds__(128)
dyn_conv(const float* __restrict__ x, const float* __restrict__ ks,
         float* __restrict__ out) {
    extern __shared__ float smem[];
    float* wl  = smem;                 // [KTOT] generated weights/biases
    float* stg = smem + KTOT;          // [4][STG_PER_WAVE]

    const int b   = blockIdx.y;
    const int tid = threadIdx.x;

    // stage this sample's generated parameters into LDS
    const float* __restrict__ ksb = ks + (size_t)b * KTOT;
    for (int i = tid; i < KTOT; i += 128) wl[i] = ksb[i];
    __syncthreads();

    const int wave = tid >> 5;
    const int lane = tid & 31;
    const int n    = lane & 15;
    const int half = lane >> 4;
    const int kk   = half * 2;
    const int col0 = (blockIdx.x * 4 + wave) * 16;

    // ---- x B-fragments (FIN=32 -> 8 k-steps), reused by layer1 & shortcut
    // unsigned element offsets from uniform per-sample base (fits 24-bit
    // immediates per channel: 64KB stride)
    const float* __restrict__ xb = x + (size_t)b * FIN * HWPX;
    const unsigned xl = (unsigned)kk * HWPX + (unsigned)(col0 + n);
    v2f xf[8];
#pragma unroll
    for (int kb = 0; kb < 8; ++kb) {
        xf[kb].x = xb[xl + (unsigned)(kb * 4) * HWPX];
        xf[kb].y = xb[xl + (unsigned)(kb * 4 + 1) * HWPX];
    }

    float* st = stg + wave * STG_PER_WAVE;

    // ---- layer 1: y1[64,16] = k_in[64,32] @ x[32,16] + b_in, leaky
    v8f acc[4];
#pragma unroll
    for (int mt = 0; mt < 4; ++mt)
#pragma unroll
        for (int r = 0; r < 8; ++r)
            acc[mt][r] = wl[OFF_BIN + mt * 16 + half * 8 + r];
#pragma unroll
    for (int kb = 0; kb < 8; ++kb) {
        const int k0 = kb * 4 + kk;
#pragma unroll
        for (int mt = 0; mt < 4; ++mt) {
            const int m = mt * 16 + n;
            v2f a; a.x = wl[OFF_KIN + m * FIN + k0];
                   a.y = wl[OFF_KIN + m * FIN + k0 + 1];
            acc[mt] = wmma4(a, xf[kb], acc[mt]);
        }
    }
#pragma unroll
    for (int mt = 0; mt < 4; ++mt)
#pragma unroll
        for (int r = 0; r < 8; ++r) {
            float v = acc[mt][r];
            v = v > 0.0f ? v : 0.01f * v;
            st[n * STG_STRIDE + mt * 16 + half * 8 + r] = v;   // transposed [n][m]
        }

    // ---- layer 2: y2[64,16] = k_mid[64,64] @ y1 + b_mid, leaky
    v8f acc2[4];
#pragma unroll
    for (int mt = 0; mt < 4; ++mt)
#pragma unroll
        for (int r = 0; r < 8; ++r)
            acc2[mt][r] = wl[OFF_BMID + mt * 16 + half * 8 + r];
#pragma unroll
    for (int kb = 0; kb < 16; ++kb) {
        const int k0 = kb * 4 + kk;
        v2f bf; bf.x = st[n * STG_STRIDE + k0];
                bf.y = st[n * STG_STRIDE + k0 + 1];
#pragma unroll
        for (int mt = 0; mt < 4; ++mt) {
            const int m = mt * 16 + n;
            v2f a; a.x = wl[OFF_KMID + m * FH + k0];
                   a.y = wl[OFF_KMID + m * FH + k0 + 1];
            acc2[mt] = wmma4(a, bf, acc2[mt]);
        }
    }
#pragma unroll
    for (int mt = 0; mt < 4; ++mt)
#pragma unroll
        for (int r = 0; r < 8; ++r) {
            float v = acc2[mt][r];
            v = v > 0.0f ? v : 0.01f * v;
            st[n * STG_STRIDE + mt * 16 + half * 8 + r] = v;
        }

    // ---- layer 3: y3[32,16] = k_out[32,64] @ y2 + b_out
    v8f acc3[2];
#pragma unroll
    for (int mt = 0; mt < 2; ++mt)
#pragma unroll
        for (int r = 0; r < 8; ++r)
            acc3[mt][r] = wl[OFF_BOUT + mt * 16 + half * 8 + r];
#pragma unroll
    for (int kb = 0; kb < 16; ++kb) {
        const int k0 = kb * 4 + kk;
        v2f bf; bf.x = st[n * STG_STRIDE + k0];
                bf.y = st[n * STG_STRIDE + k0 + 1];
#pragma unroll
        for (int mt = 0; mt < 2; ++mt) {
            const int m = mt * 16 + n;
            v2f a; a.x = wl[OFF_KOUT + m * FH + k0];
                   a.y = wl[OFF_KOUT + m * FH + k0 + 1];
            acc3[mt] = wmma4(a, bf, acc3[mt]);
        }
    }

    // ---- shortcut: xs[32,16] = k_short[32,32] @ x + b_short (reuses xf)
    v8f accs[2];
#pragma unroll
    for (int mt = 0; mt < 2; ++mt)
#pragma unroll
        for (int r = 0; r < 8; ++r)
            accs[mt][r] = wl[OFF_BSH + mt * 16 + half * 8 + r];
#pragma unroll
    for (int kb = 0; kb < 8; ++kb) {
        const int k0 = kb * 4 + kk;
#pragma unroll
        for (int mt = 0; mt < 2; ++mt) {
            const int m = mt * 16 + n;
            v2f a; a.x = wl[OFF_KSH + m * FIN + k0];
                   a.y = wl[OFF_KSH + m * FIN + k0 + 1];
            accs[mt] = wmma4(a, xf[kb], accs[mt]);
        }
    }

    // ---- out = y3 + xs  (unsigned offsets -> base + 24-bit immediates)
    float* __restrict__ ob = out + (size_t)b * FOUT * HWPX;
    const unsigned ol = (unsigned)(half * 8) * HWPX + (unsigned)(col0 + n);
#pragma unroll
    for (int mt = 0; mt < 2; ++mt)
#pragma unroll
        for (int r = 0; r < 8; ++r) {
            const unsigned m = (unsigned)(mt * 16 + r) * HWPX;
            ob[ol + m] = acc3[mt][r] + accs[mt][r];
        }
}

// ---------------------------------------------------------------------------
// Host launcher
// ---------------------------------------------------------------------------
extern "C" void kernel_launch(void* const* d_in, const int* in_sizes, int n_in,
                              void* d_out, int out_size, void* d_ws, size_t ws_size,
                              hipStream_t stream) {
    const float* x   = (const float*)d_in[0];
    const float* lat = (const float*)d_in[1];
    const float* W1  = (const float*)d_in[2];
    const float* b1  = (const float*)d_in[3];
    const float* W2  = (const float*)d_in[4];
    const float* b2  = (const float*)d_in[5];
    float* out = (float*)d_out;

    float* h     = (float*)d_ws;                       // [32, 9408]
    float* ksbuf = h + (size_t)NB * KTOT;              // [32, 9408]

    // N=9408 -> 588 col tiles -> 147 blocks of 4 waves (exact)
    hyper_gemm<<<147, 128, 0, stream>>>(lat, W1, b1, h,     LATN, KTOT);
    hyper_gemm<<<147, 128, 0, stream>>>(h,   W2, b2, ksbuf, KTOT, KTOT);

    dim3 grid(HWPX / 64, NB);                          // 256 x 32 blocks
    size_t shmem = (size_t)(KTOT + 4 * STG_PER_WAVE) * sizeof(float); // 54,272 B
    dyn_conv<<<grid, 128, shmem, stream>>>(x, ksbuf, out);
}